// NodeModule_4432406249918
// MI455X (gfx1250) — compile-verified
//
#include <hip/hip_runtime.h>

typedef __attribute__((ext_vector_type(2))) float v2f;
typedef __attribute__((ext_vector_type(4))) float v4f;
typedef __attribute__((ext_vector_type(8))) float v8f;

#define NV    4096            // n_vars
#define NC    4097            // n_vars + 1 (columns / coeff count)
#define KTOT  4096            // reduction length (K)
#define LDM   4097            // row stride of layer_inputs (K, n_vars+1)
#define SLICES 16             // K split
#define KSL   (KTOT / SLICES) // 256 rows per slice
#define TILES 257             // ceil(NC / 16) column tiles
#define PCOLS (TILES * 16)    // 4112 padded partial columns

// ---------------------------------------------------------------------------
// Stage 1: dual GEMV on the matrix pipe.
// A (16xK) rows: row0 = pos, row1 = neg, rows 2..15 = 0.
// C_u += A x B_under ; C_o += A x B_over  via V_WMMA_F32_16X16X4_F32.
// Row 0 of C_u = pos.under, row 1 = neg.under (same for C_o with over).
// Partials {pu, nu, po, no} per (slice, col) go to scratch inside d_out.
// ---------------------------------------------------------------------------
__global__ __launch_bounds__(32)
void gemv_wmma_stage1(const float* __restrict__ under,
                      const float* __restrict__ over,
                      const float* __restrict__ pos,
                      const float* __restrict__ neg,
                      v4f* __restrict__ partials)
{
    __shared__ float wpos[KSL];
    __shared__ float wneg[KSL];

    const int tile  = blockIdx.x;     // 0..TILES-1
    const int slice = blockIdx.y;     // 0..SLICES-1
    const int lane  = threadIdx.x;    // 0..31 (wave32)
    const int k0    = slice * KSL;

    // Stage the weight slice into LDS (shared by the whole wave's K loop).
    for (int i = lane; i < KSL; i += 32) {
        wpos[i] = pos[k0 + i];
        wneg[i] = neg[k0 + i];
    }
    __syncthreads();

    const int m  = lane & 15;   // M row (A) / N col (B) within the tile
    const int hi = lane >> 4;   // half-wave selects K sub-pair
    const int j  = tile * 16 + m;              // output column
    const int jc = (j < NC) ? j : (NC - 1);    // clamped load column

    v8f cu = {};
    v8f co = {};

    for (int kk = 0; kk < KSL; kk += 4) {
        const int ka = kk + 2 * hi;            // K offset for this half-wave
        // A fragment: 16x4 f32 — lane m holds W[m][ka], W[m][ka+1]
        const float ax = (m == 0) ? wpos[ka]     : (m == 1) ? wneg[ka]     : 0.0f;
        const float ay = (m == 0) ? wpos[ka + 1] : (m == 1) ? wneg[ka + 1] : 0.0f;
        v2f a; a.x = ax; a.y = ay;

        // B fragment: 4x16 f32 — lane n holds X[k0+ka][j], X[k0+ka+1][j]
        const long rowb = (long)(k0 + ka) * LDM + jc;
        v2f bu, bo;
        bu.x = __builtin_nontemporal_load(under + rowb);
        bu.y = __builtin_nontemporal_load(under + rowb + LDM);
        bo.x = __builtin_nontemporal_load(over  + rowb);
        bo.y = __builtin_nontemporal_load(over  + rowb + LDM);

        cu = __builtin_amdgcn_wmma_f32_16x16x4_f32(false, a, false, bu,
                                                   (short)0, cu, false, false);
        co = __builtin_amdgcn_wmma_f32_16x16x4_f32(false, a, false, bo,
                                                   (short)0, co, false, false);
    }

    // C/D layout: VGPR r, lanes 0-15 -> M=r, N=lane. Rows 0/1 live in cu[0],cu[1].
    if (hi == 0 && j < NC) {
        v4f p; p.x = cu[0]; p.y = cu[1]; p.z = co[0]; p.w = co[1];
        partials[(long)slice * PCOLS + j] = p;
    }
}

// ---------------------------------------------------------------------------
// Stage 2: deterministic fixed-order reduction over the K slices + bias.
//   under_coeffs[j] = neg.over + pos.under + bias = no + pu + b
//   over_coeffs[j]  = pos.over + neg.under + bias = po + nu + b
// ---------------------------------------------------------------------------
__global__ void reduce_stage2(const v4f* __restrict__ partials,
                              const float* __restrict__ bias,
                              float* __restrict__ ucoef,
                              float* __restrict__ ocoef)
{
    const int jcol = blockIdx.x * blockDim.x + threadIdx.x;
    if (jcol >= NC) return;
    float pu = 0.f, nu = 0.f, po = 0.f, no = 0.f;
    for (int s = 0; s < SLICES; ++s) {
        const v4f p = partials[(long)s * PCOLS + jcol];
        pu += p.x; nu += p.y; po += p.z; no += p.w;
    }
    const float b = bias[0];
    ucoef[jcol] = no + pu + b;
    ocoef[jcol] = po + nu + b;
}

// ---------------------------------------------------------------------------
// Stage 3: structured IBF fill (write-bound). Each thread emits one v4f
// (two rows x 2 interval cols) for BOTH outputs, with non-temporal stores
// (outputs are larger than L2 and never re-read).
// Block bb, row r, col c:
//   r==0        : coeffs[bb] * (bb==0 ? intervals[0,c] : 1)
//   r==bb (>0)  : intervals[r,c]
//   else        : 1
// ---------------------------------------------------------------------------
__global__ void fill_stage3(const float* __restrict__ ucoef,
                            const float* __restrict__ ocoef,
                            const float* __restrict__ intervals,
                            float* __restrict__ out)
{
    const long PAIRS = NV / 2;                  // 2048 row-pairs per block
    const long gid = (long)blockIdx.x * blockDim.x + threadIdx.x;
    if (gid >= (long)NC * PAIRS) return;

    const int bb = (int)(gid / PAIRS);
    const int rp = (int)(gid % PAIRS);
    const int r0 = rp * 2;                      // even row; r1 = r0 + 1

    v4f u; u.x = 1.f; u.y = 1.f; u.z = 1.f; u.w = 1.f;
    v4f o = u;

    if (r0 == 0) {
        const float uc = ucoef[bb], oc = ocoef[bb];
        if (bb == 0) {
            const float i0 = intervals[0], i1 = intervals[1];
            u.x = uc * i0; u.y = uc * i1;
            o.x = oc * i0; o.y = oc * i1;
        } else {
            u.x = uc; u.y = uc;
            o.x = oc; o.y = oc;
        }
        if (bb == 1) {                          // row 1 of block 1 is diagonal
            const float i0 = intervals[2], i1 = intervals[3];
            u.z = i0; u.w = i1; o.z = i0; o.w = i1;
        }
    } else {
        if (bb == r0) {
            const float i0 = intervals[2 * r0], i1 = intervals[2 * r0 + 1];
            u.x = i0; u.y = i1; o.x = i0; o.y = i1;
        }
        if (bb == r0 + 1) {
            const float i0 = intervals[2 * (r0 + 1)], i1 = intervals[2 * (r0 + 1) + 1];
            u.z = i0; u.w = i1; o.z = i0; o.w = i1;
        }
    }

    const long base = ((long)bb * NV + r0) * 2;                  // float index
    __builtin_nontemporal_store(u, (v4f*)(out + base));
    __builtin_nontemporal_store(o, (v4f*)(out + (long)NC * NV * 2 + base));
}

// Tiny tail: both degree vectors are ones(n_vars).
__global__ void degrees_stage4(float* __restrict__ out)
{
    const int i = blockIdx.x * blockDim.x + threadIdx.x;
    if (i < 2 * NV) {
        out[(long)2 * NC * NV * 2 + i] = 1.0f;
    }
}

extern "C" void kernel_launch(void* const* d_in, const int* in_sizes, int n_in,
                              void* d_out, int out_size, void* d_ws, size_t ws_size,
                              hipStream_t stream)
{
    (void)in_sizes; (void)n_in; (void)out_size; (void)ws_size;

    const float* under     = (const float*)d_in[0];  // (K, NC)
    const float* over      = (const float*)d_in[1];  // (K, NC)
    /* d_in[2], d_in[3]: degree vectors (all ones, outputs fixed) */
    const float* intervals = (const float*)d_in[4];  // (NV, 2)
    const float* pos       = (const float*)d_in[5];  // (K,)
    const float* neg       = (const float*)d_in[6];  // (K,)
    const float* bias      = (const float*)d_in[7];  // (1,)

    float* out = (float*)d_out;
    // Partials (~1 MB) staged in d_out; fully consumed by stage 2 before
    // stage 3 overwrites the whole output buffer. ws holds only the coeffs.
    v4f*   partials = (v4f*)d_out;
    float* ucoef    = (float*)d_ws;
    float* ocoef    = ucoef + NC;

    gemv_wmma_stage1<<<dim3(TILES, SLICES), 32, 0, stream>>>(under, over, pos, neg, partials);
    reduce_stage2<<<(NC + 255) / 256, 256, 0, stream>>>(partials, bias, ucoef, ocoef);

    const long tot = (long)NC * (NV / 2);
    fill_stage3<<<(unsigned)((tot + 255) / 256), 256, 0, stream>>>(ucoef, ocoef, intervals, out);
    degrees_stage4<<<(2 * NV + 255) / 256, 256, 0, stream>>>(out);
}